// GNN_2L_int_no_edge_type_25125558682224
// MI455X (gfx1250) — compile-verified
//
#include <hip/hip_runtime.h>
#include <hip/hip_bf16.h>

typedef __attribute__((ext_vector_type(2))) float v2f;
typedef __attribute__((ext_vector_type(8))) float v8f;

#define D 64           // feature width throughout conv stack (D_IN == HID == P == 64)
#define WAVES_PER_BLOCK 8

// ---------------------------------------------------------------------------
// zero fill
// ---------------------------------------------------------------------------
__global__ void zero_kernel(float* __restrict__ p, long long n) {
    long long i = (long long)blockIdx.x * blockDim.x + threadIdx.x;
    if (i < n) p[i] = 0.0f;
}

// ---------------------------------------------------------------------------
// h0 = concat(x[N,32], FT[x_ft_index][N,32]) -> [N,64]; float4 granularity
// ---------------------------------------------------------------------------
__global__ void build_h_kernel(const float* __restrict__ x,
                               const float* __restrict__ FT,
                               const int*   __restrict__ idx,
                               float* __restrict__ h0, int n_nodes) {
    long long tid = (long long)blockIdx.x * blockDim.x + threadIdx.x;
    long long total = (long long)n_nodes * 16;      // 16 float4 per node row (64 floats)
    if (tid >= total) return;
    int node = (int)(tid >> 4);
    int q    = (int)(tid & 15);
    float4 v;
    if (q < 8) {
        v = ((const float4*)(x + (size_t)node * 32))[q];
    } else {
        v = ((const float4*)(FT + (size_t)idx[node] * 32))[q - 8];
    }
    ((float4*)(h0 + (size_t)node * D))[q] = v;
}

// ---------------------------------------------------------------------------
// agg[dst] += h[src] over all edges; 64 threads per edge, fp32 HW atomics
// ---------------------------------------------------------------------------
__global__ void scatter_add_kernel(const float* __restrict__ h,
                                   const int* __restrict__ src,
                                   const int* __restrict__ dst,
                                   float* __restrict__ agg, long long total) {
    long long t = (long long)blockIdx.x * blockDim.x + threadIdx.x;
    if (t >= total) return;
    int e = (int)(t >> 6);
    int c = (int)(t & 63);
    int s = src[e];
    int d = dst[e];
    atomicAdd(&agg[(size_t)d * D + c], h[(size_t)s * D + c]);
}

// ---------------------------------------------------------------------------
// out = leaky(agg @ Wl + h @ Ws + b)   shapes: [N,64]@[64,64]
// One wave per 16-row strip; 4 column tiles of 16; K marched in steps of 4
// via v_wmma_f32_16x16x4_f32 (fp32 in, fp32 acc — matches reference math).
// ---------------------------------------------------------------------------
__global__ __launch_bounds__(256) void conv_wmma_kernel(
        const float* __restrict__ agg, const float* __restrict__ h,
        const float* __restrict__ Wl,  const float* __restrict__ Ws,
        const float* __restrict__ bias, float* __restrict__ out, int n_rows) {
    const int wave = threadIdx.x >> 5;
    const int lane = threadIdx.x & 31;
    const int row_base = (blockIdx.x * WAVES_PER_BLOCK + wave) * 16;
    if (row_base >= n_rows) return;          // wave-uniform: EXEC stays all-1s
    const int half = lane >> 4;              // 0: lanes 0-15, 1: lanes 16-31
    const int l16  = lane & 15;
    const int row  = row_base + l16;         // A-matrix M index

    #pragma unroll
    for (int nt = 0; nt < 4; ++nt) {
        const int n0 = nt * 16;
        const float bv = bias[n0 + l16];
        v8f c;
        #pragma unroll
        for (int v = 0; v < 8; ++v) c[v] = bv;

        #pragma unroll
        for (int k0 = 0; k0 < D; k0 += 4) {
            const int ka = k0 + 2 * half;    // this half-wave's K pair
            // A fragments: contiguous 64-bit loads from the row
            v2f aA = *(const v2f*)(agg + (size_t)row * D + ka);
            v2f aH = *(const v2f*)(h   + (size_t)row * D + ka);
            // B fragments: W[k][n] row-major, column n0+l16
            v2f bL, bS;
            bL.x = Wl[(size_t)(ka + 0) * D + n0 + l16];
            bL.y = Wl[(size_t)(ka + 1) * D + n0 + l16];
            bS.x = Ws[(size_t)(ka + 0) * D + n0 + l16];
            bS.y = Ws[(size_t)(ka + 1) * D + n0 + l16];
            c = __builtin_amdgcn_wmma_f32_16x16x4_f32(false, aA, false, bL,
                                                      (short)0, c, false, false);
            c = __builtin_amdgcn_wmma_f32_16x16x4_f32(false, aH, false, bS,
                                                      (short)0, c, false, false);
        }

        #pragma unroll
        for (int v = 0; v < 8; ++v) {
            float val = c[v];
            val = (val >= 0.0f) ? val : 0.1f * val;     // leaky(0.1)
            out[(size_t)(row_base + v + 8 * half) * D + n0 + l16] = val;
        }
    }
}

// ---------------------------------------------------------------------------
// p = h @ Wp then g[batch[row]] += p[row]  (fused projection + add-pool)
// ---------------------------------------------------------------------------
__global__ __launch_bounds__(256) void proj_pool_wmma_kernel(
        const float* __restrict__ h, const float* __restrict__ Wp,
        const int* __restrict__ batch, float* __restrict__ g, int n_rows) {
    const int wave = threadIdx.x >> 5;
    const int lane = threadIdx.x & 31;
    const int row_base = (blockIdx.x * WAVES_PER_BLOCK + wave) * 16;
    if (row_base >= n_rows) return;
    const int half = lane >> 4;
    const int l16  = lane & 15;
    const int row  = row_base + l16;

    #pragma unroll
    for (int nt = 0; nt < 4; ++nt) {
        const int n0 = nt * 16;
        v8f c = {};
        #pragma unroll
        for (int k0 = 0; k0 < D; k0 += 4) {
            const int ka = k0 + 2 * half;
            v2f a = *(const v2f*)(h + (size_t)row * D + ka);
            v2f b;
            b.x = Wp[(size_t)(ka + 0) * D + n0 + l16];
            b.y = Wp[(size_t)(ka + 1) * D + n0 + l16];
            c = __builtin_amdgcn_wmma_f32_16x16x4_f32(false, a, false, b,
                                                      (short)0, c, false, false);
        }
        #pragma unroll
        for (int v = 0; v < 8; ++v) {
            const int r = row_base + v + 8 * half;
            atomicAdd(&g[(size_t)batch[r] * D + n0 + l16], c[v]);
        }
    }
}

// ---------------------------------------------------------------------------
// per-graph ANN head: leaky(g) -> relu(@A0+b0) -> relu(@A1+b1) -> @A2+b2
// one 128-thread block per graph (64->128->32->1, 25 MFLOP total)
// ---------------------------------------------------------------------------
__global__ __launch_bounds__(128) void ann_kernel(
        const float* __restrict__ g,
        const float* __restrict__ A0W, const float* __restrict__ A0b,
        const float* __restrict__ A1W, const float* __restrict__ A1b,
        const float* __restrict__ A2W, const float* __restrict__ A2b,
        float* __restrict__ out) {
    __shared__ float sg[64];
    __shared__ float s1[128];
    __shared__ float s2[32];
    const int gid = blockIdx.x;
    const int t   = threadIdx.x;

    if (t < 64) {
        float v = g[(size_t)gid * D + t];
        sg[t] = (v >= 0.0f) ? v : 0.1f * v;     // leaky on pooled features
    }
    __syncthreads();

    // layer 0: 64 -> 128
    float acc = A0b[t];
    #pragma unroll 8
    for (int k = 0; k < 64; ++k) acc += sg[k] * A0W[(size_t)k * 128 + t];
    s1[t] = (acc > 0.0f) ? acc : 0.0f;
    __syncthreads();

    // layer 1: 128 -> 32
    if (t < 32) {
        float a = A1b[t];
        #pragma unroll 8
        for (int k = 0; k < 128; ++k) a += s1[k] * A1W[(size_t)k * 32 + t];
        s2[t] = (a > 0.0f) ? a : 0.0f;
    }
    __syncthreads();

    // layer 2: 32 -> 1
    if (t == 0) {
        float a = A2b[0];
        #pragma unroll
        for (int k = 0; k < 32; ++k) a += s2[k] * A2W[k];
        out[gid] = a;
    }
}

// ---------------------------------------------------------------------------
extern "C" void kernel_launch(void* const* d_in, const int* in_sizes, int n_in,
                              void* d_out, int out_size, void* d_ws, size_t ws_size,
                              hipStream_t stream) {
    const float* x   = (const float*)d_in[0];
    const float* FT  = (const float*)d_in[1];
    const float* W0l = (const float*)d_in[2];
    const float* W0s = (const float*)d_in[3];
    const float* b0  = (const float*)d_in[4];
    const float* W1l = (const float*)d_in[5];
    const float* W1s = (const float*)d_in[6];
    const float* b1  = (const float*)d_in[7];
    const float* Wp  = (const float*)d_in[8];
    const float* A0W = (const float*)d_in[9];
    const float* A0b = (const float*)d_in[10];
    const float* A1W = (const float*)d_in[11];
    const float* A1b = (const float*)d_in[12];
    const float* A2W = (const float*)d_in[13];
    const float* A2b = (const float*)d_in[14];
    const int* xft   = (const int*)d_in[15];
    const int* eidx  = (const int*)d_in[16];
    const int* batch = (const int*)d_in[17];

    const int N = in_sizes[0] / 32;           // 100000
    const int E = in_sizes[16] / 2;           // 1600000
    const int G = out_size;                   // 2048
    const int* e_src = eidx;
    const int* e_dst = eidx + E;

    // workspace layout (floats)
    float* hA  = (float*)d_ws;                // [N,64]
    float* agg = hA  + (size_t)N * D;         // [N,64]
    float* hB  = agg + (size_t)N * D;         // [N,64]
    float* g   = hB  + (size_t)N * D;         // [G,64]

    const int TB = 256;
    const long long nh   = (long long)N * 16;           // build_h float4 units
    const long long nsc  = (long long)E * 64;           // scatter lanes
    const long long nzA  = (long long)N * D;
    const long long nzG  = (long long)G * D;
    const int grid_h   = (int)((nh  + TB - 1) / TB);
    const int grid_sc  = (int)((nsc + TB - 1) / TB);
    const int grid_zA  = (int)((nzA + TB - 1) / TB);
    const int grid_zG  = (int)((nzG + TB - 1) / TB);
    const int grid_mm  = (N + 16 * WAVES_PER_BLOCK - 1) / (16 * WAVES_PER_BLOCK);

    // h0 = [x | FT[idx]]
    build_h_kernel<<<grid_h, TB, 0, stream>>>(x, FT, xft, hA, N);

    // conv 0
    zero_kernel<<<grid_zA, TB, 0, stream>>>(agg, nzA);
    scatter_add_kernel<<<grid_sc, TB, 0, stream>>>(hA, e_src, e_dst, agg, nsc);
    conv_wmma_kernel<<<grid_mm, TB, 0, stream>>>(agg, hA, W0l, W0s, b0, hB, N);

    // conv 1
    zero_kernel<<<grid_zA, TB, 0, stream>>>(agg, nzA);
    scatter_add_kernel<<<grid_sc, TB, 0, stream>>>(hB, e_src, e_dst, agg, nsc);
    conv_wmma_kernel<<<grid_mm, TB, 0, stream>>>(agg, hB, W1l, W1s, b1, hA, N);

    // projection + global add pool
    zero_kernel<<<grid_zG, TB, 0, stream>>>(g, nzG);
    proj_pool_wmma_kernel<<<grid_mm, TB, 0, stream>>>(hA, Wp, batch, g, N);

    // ANN head
    ann_kernel<<<G, 128, 0, stream>>>(g, A0W, A0b, A1W, A1b, A2W, A2b,
                                      (float*)d_out);
}